// DirectionalSeparableConv2D_37701222924350
// MI455X (gfx1250) — compile-verified
//
#include <hip/hip_runtime.h>

// ---------------------------------------------------------------------------
// DirectionalSeparableConv2D fused kernel for MI455X (gfx1250, wave32, WMMA)
//
// Roofline: 302 MB min HBM traffic (~13us @ 23.3 TB/s) vs ~10 GFLOP of f32
// matrix math -> memory bound. Fuse depthwise conv + 1x1 mixing; mixing is a
// dense 128x128 GEMM per pixel done with V_WMMA_F32_16X16X4_F32 (f32 keeps
// full precision and costs nothing extra at this arithmetic intensity).
// ---------------------------------------------------------------------------

#define B_    128
#define H_    48
#define W_    48
#define CIN   128
#define COUT  128
#define CEN   32
#define DIRC  24
#define ROWS  4                  // output rows per workgroup
#define NPIX  (ROWS * W_)        // 192 pixels per workgroup
#define YSTR  208                // padded LDS stride (16-bank group offset => conflict-free)
#define NGRP  (NPIX / 16)        // 12 pixel groups of 16

typedef float v2f __attribute__((ext_vector_type(2)));
typedef float v8f __attribute__((ext_vector_type(8)));

// Build the dense 128x128 mixing matrix Wbig from the 5 block matrices.
// out rows: [0,32)=cen, [32,56)=h, [56,80)=v, [80,104)=d1, [104,128)=d2
// y cols follow the same channel order as the input split.
__global__ __launch_bounds__(256) void build_wbig_kernel(
    const float* __restrict__ c2c,   // (32,32)
    const float* __restrict__ p2c,   // (32,24)
    const float* __restrict__ d2c,   // (32,24)
    const float* __restrict__ c2d,   // (24,32)
    const float* __restrict__ d2d,   // (24,24)
    float* __restrict__ wbig)        // (128,128)
{
    int i = blockIdx.x * blockDim.x + threadIdx.x;
    if (i >= COUT * CIN) return;
    int o = i >> 7;
    int c = i & 127;
    float v = 0.0f;
    if (o < CEN) {
        if (c < CEN)        v = c2c[o * CEN + c];
        else if (c < 80)    v = p2c[o * DIRC + ((c - 32) % DIRC)];   // yh + yv share par2cen
        else                v = d2c[o * DIRC + ((c - 80) % DIRC)];   // yd1 + yd2 share dia2cen
    } else {
        int blk = (o - 32) / DIRC;          // which directional output block
        int oo  = (o - 32) % DIRC;
        if (c < CEN) {
            v = c2d[oo * CEN + c];          // od_cen added to every directional block
        } else {
            int cblk = (c - 32) / DIRC;
            int cc   = (c - 32) % DIRC;
            v = (cblk == blk) ? d2d[oo * DIRC + cc] : 0.0f;
        }
    }
    wbig[i] = v;
}

__device__ __forceinline__ float ldx(const float* __restrict__ xb, int hh, int ww) {
    return (hh >= 0 && hh < H_ && ww >= 0 && ww < W_) ? xb[hh * W_ + ww] : 0.0f;
}

__global__ __launch_bounds__(256) void dsconv_fused_kernel(
    const float* __restrict__ x,      // (B,128,48,48)
    const float* __restrict__ wcen,   // (32,3,3)
    const float* __restrict__ wdir,   // (24,5)
    const float* __restrict__ wbig,   // (128,128)
    float* __restrict__ out)          // (B,128,48,48)
{
    __shared__ float s_ys[CIN * YSTR];      // y[ch][pix], padded stride
    __shared__ float s_wcen[CEN * 9];
    __shared__ float s_wdir[DIRC * 5];

    const int tid  = threadIdx.x;
    const int b    = blockIdx.y;
    const int r0   = blockIdx.x * ROWS;

    const int lane = tid & 31;
    const int wv   = tid >> 5;        // wave id 0..7 -> out-channel tile (16 ch)
    const int m    = lane & 15;       // A-frag row / B-frag column / D column
    const int kh   = lane >> 4;       // lane half

    // ---- Preload this wave's 16x128 slice of Wbig as 32 A-fragments (regs).
    // A 16x4 f32 layout: lanes0-15 M=0..15; VGPR0 holds K=2*kh, VGPR1 K=2*kh+1.
    v2f aF[32];
#pragma unroll
    for (int kb = 0; kb < 32; ++kb) {
        const float* src = wbig + (wv * 16 + m) * CIN + kb * 4 + kh * 2;
        aF[kb] = *(const v2f*)src;
    }

    // ---- Stage the tiny depthwise kernels in LDS.
    for (int i = tid; i < CEN * 9; i += 256)  s_wcen[i] = wcen[i];
    for (int i = tid; i < DIRC * 5; i += 256) s_wdir[i] = wdir[i];
    __syncthreads();

    // ---- Phase A: depthwise conv, global -> LDS y tile (128 ch x 192 px).
    for (int i = tid; i < CIN * NPIX; i += 256) {
        int c   = i / NPIX;
        int p   = i - c * NPIX;
        int r   = p / W_;
        int col = p - r * W_;
        int h   = r0 + r;
        const float* xb = x + ((size_t)(b * CIN + c)) * (H_ * W_);
        float acc = 0.0f;
        if (c < CEN) {
            const float* kw = &s_wcen[c * 9];
#pragma unroll
            for (int ky = 0; ky < 3; ++ky)
#pragma unroll
                for (int kx = 0; kx < 3; ++kx)
                    acc += kw[ky * 3 + kx] * ldx(xb, h + ky - 1, col + kx - 1);
        } else {
            int t   = c - CEN;
            int cls = t / DIRC;                 // 0=h 1=v 2=diag 3=anti-diag
            int cc  = t - cls * DIRC;
            const float* kw = &s_wdir[cc * 5];
#pragma unroll
            for (int j = 0; j < 5; ++j) {
                int dy = (cls == 0) ? 0 : (j - 2);
                int dx = (cls == 1) ? 0 : ((cls == 3) ? (2 - j) : (j - 2));
                acc += kw[j] * ldx(xb, h + dy, col + dx);
            }
        }
        s_ys[c * YSTR + p] = acc;
    }
    __syncthreads();

    // ---- Phase B: Out(16x16) = Wbig_tile(16x128) @ Y(128x16) via 32 chained
    // v_wmma_f32_16x16x4_f32, two accumulators for ILP. EXEC uniform here.
    for (int g = 0; g < NGRP; ++g) {
        v8f c0 = {};
        v8f c1 = {};
        const int pb = g * 16 + m;
#pragma unroll
        for (int kb = 0; kb < 32; kb += 2) {
            int k0 = kb * 4 + kh * 2;
            // B 4x16 f32 layout: VGPR0 lanes0-15 K=k0, lanes16-31 K=k0+2 (k0
            // already folds in kh*2); VGPR1 is K=k0+1. Rows k and k+1 land in
            // disjoint bank groups thanks to YSTR=208.
            v2f b0, b1;
            b0.x = s_ys[(k0 + 0) * YSTR + pb];
            b0.y = s_ys[(k0 + 1) * YSTR + pb];
            b1.x = s_ys[(k0 + 4) * YSTR + pb];
            b1.y = s_ys[(k0 + 5) * YSTR + pb];
            c0 = __builtin_amdgcn_wmma_f32_16x16x4_f32(
                false, aF[kb],     false, b0, (short)0, c0, false, false);
            c1 = __builtin_amdgcn_wmma_f32_16x16x4_f32(
                false, aF[kb + 1], false, b1, (short)0, c1, false, false);
        }
        v8f cs = c0 + c1;
        // D layout: VGPR v holds M = v + 8*kh, N = lane&15.
        const int r  = g / 3;
        const int cb = (g % 3) * 16 + m;
        float* ob = out + (((size_t)b * COUT + wv * 16 + kh * 8) * H_ + (r0 + r)) * W_ + cb;
#pragma unroll
        for (int v = 0; v < 8; ++v)
            ob[(size_t)v * (H_ * W_)] = cs[v];
    }
}

extern "C" void kernel_launch(void* const* d_in, const int* in_sizes, int n_in,
                              void* d_out, int out_size, void* d_ws, size_t ws_size,
                              hipStream_t stream) {
    (void)in_sizes; (void)n_in; (void)out_size; (void)ws_size;
    const float* x    = (const float*)d_in[0];
    const float* cent = (const float*)d_in[1];
    const float* dirt = (const float*)d_in[2];
    const float* c2c  = (const float*)d_in[3];
    const float* p2c  = (const float*)d_in[4];
    const float* d2c  = (const float*)d_in[5];
    const float* c2d  = (const float*)d_in[6];
    const float* d2d  = (const float*)d_in[7];
    float* out  = (float*)d_out;
    float* wbig = (float*)d_ws;   // 128*128*4 = 64 KB scratch

    build_wbig_kernel<<<(COUT * CIN + 255) / 256, 256, 0, stream>>>(
        c2c, p2c, d2c, c2d, d2d, wbig);

    dim3 grid(H_ / ROWS, B_);     // 12 row-blocks x 128 batches
    dsconv_fused_kernel<<<grid, 256, 0, stream>>>(x, cent, dirt, wbig, out);
}